// GCN_9234179686680
// MI455X (gfx1250) — compile-verified
//
#include <hip/hip_runtime.h>

typedef __attribute__((ext_vector_type(2))) float v2f;
typedef __attribute__((ext_vector_type(8))) float v8f;

#define NNODES 50000
#define NEDGES 800000
#define NF     128     // NFEAT == 2*NHID
#define NH     64      // NHID

// ---------------------------------------------------------------- degrees
__global__ void deg_kernel(const int* __restrict__ src, const int* __restrict__ dst,
                           float* __restrict__ degOut, float* __restrict__ degIn, int nE) {
    int e = blockIdx.x * blockDim.x + threadIdx.x;
    if (e < nE) {
        atomicAdd(&degOut[src[e]], 1.0f);
        atomicAdd(&degIn[dst[e]], 1.0f);
    }
}

__global__ void rscale_kernel(float* __restrict__ d, int n) {
    int i = blockIdx.x * blockDim.x + threadIdx.x;
    if (i < n) {
        float v = fmaxf(d[i], 1.0f);
        d[i] = 1.0f / sqrtf(v);
    }
}

// --------------------------------------------- h = feat * scale_out[row]  (128 cols)
__global__ void scale_rows_kernel(const float* __restrict__ x, const float* __restrict__ s,
                                  float* __restrict__ y, int n) {
    int idx = blockIdx.x * blockDim.x + threadIdx.x;   // n * 32 float4 slots
    if (idx < n * 32) {
        int row = idx >> 5;
        float4 v = ((const float4*)x)[idx];
        float sc = s[row];
        v.x *= sc; v.y *= sc; v.z *= sc; v.w *= sc;
        ((float4*)y)[idx] = v;
    }
}

// --------------------------------------------- agg[dst] += h[src], 128 floats / edge
__global__ void scatter128_kernel(const float* __restrict__ h, const int* __restrict__ src,
                                  const int* __restrict__ dst, float* __restrict__ agg, int nE) {
    int idx = blockIdx.x * blockDim.x + threadIdx.x;
    int e = idx >> 5;            // 32 float4 slots per edge
    int q = idx & 31;
    if (e < nE) {
        int s = src[e], d = dst[e];
        float4 v = ((const float4*)(h + (long long)s * NF))[q];
        float* p = agg + (long long)d * NF + q * 4;
        atomicAdd(p + 0, v.x); atomicAdd(p + 1, v.y);
        atomicAdd(p + 2, v.z); atomicAdd(p + 3, v.w);
    }
}

// --------------------------------------------- agg[dst] += h[src], 64 floats / edge
__global__ void scatter64_kernel(const float* __restrict__ h, const int* __restrict__ src,
                                 const int* __restrict__ dst, float* __restrict__ agg, int nE) {
    int idx = blockIdx.x * blockDim.x + threadIdx.x;
    int e = idx >> 4;            // 16 float4 slots per edge
    int q = idx & 15;
    if (e < nE) {
        int s = src[e], d = dst[e];
        float4 v = ((const float4*)(h + (long long)s * NH))[q];
        float* p = agg + (long long)d * NH + q * 4;
        atomicAdd(p + 0, v.x); atomicAdd(p + 1, v.y);
        atomicAdd(p + 2, v.z); atomicAdd(p + 3, v.w);
    }
}

// ---------------------------------------------------------------- GEMM layer 1
// out[M,128] = relu( (A[M,128] @ B[128,128]) * sin[row] + bias[col] )
// one wave per 16x64 tile; grid.x = (M/16)*2
__global__ void gemm1_kernel(const float* __restrict__ A, const float* __restrict__ B,
                             const float* __restrict__ bias, const float* __restrict__ sin_,
                             float* __restrict__ out) {
    const int K = NF, NB = NF;
    int tile = blockIdx.x;
    int m0 = (tile >> 1) * 16;
    int n0 = (tile & 1) * 64;
    int lane = threadIdx.x;            // 0..31
    int lr   = lane & 15;
    int kh   = (lane >> 4) << 1;       // 0 (lanes 0-15) or 2 (lanes 16-31)

    v8f c0 = {}, c1 = {}, c2 = {}, c3 = {};
    const float* arow = A + (long long)(m0 + lr) * K;

    #pragma unroll 4
    for (int k = 0; k < K; k += 4) {
        v2f a;
        a.x = arow[k + kh];
        a.y = arow[k + kh + 1];
        const float* b0 = B + (long long)(k + kh) * NB + n0 + lr;
        v2f bb0, bb1, bb2, bb3;
        bb0.x = b0[0];  bb0.y = b0[NB];
        bb1.x = b0[16]; bb1.y = b0[NB + 16];
        bb2.x = b0[32]; bb2.y = b0[NB + 32];
        bb3.x = b0[48]; bb3.y = b0[NB + 48];
        c0 = __builtin_amdgcn_wmma_f32_16x16x4_f32(false, a, false, bb0, (short)0, c0, false, false);
        c1 = __builtin_amdgcn_wmma_f32_16x16x4_f32(false, a, false, bb1, (short)0, c1, false, false);
        c2 = __builtin_amdgcn_wmma_f32_16x16x4_f32(false, a, false, bb2, (short)0, c2, false, false);
        c3 = __builtin_amdgcn_wmma_f32_16x16x4_f32(false, a, false, bb3, (short)0, c3, false, false);
    }

    int rbase = m0 + ((lane >> 4) << 3);   // rows m0..m0+7 (lanes<16) or m0+8..m0+15
    int col   = lane & 15;
    #pragma unroll
    for (int r = 0; r < 8; ++r) {
        int row = rbase + r;
        float si = sin_[row];
        float* orow = out + (long long)row * NB + n0;
        float v;
        v = c0[r] * si + bias[n0 + col];      orow[col]      = fmaxf(v, 0.0f);
        v = c1[r] * si + bias[n0 + 16 + col]; orow[16 + col] = fmaxf(v, 0.0f);
        v = c2[r] * si + bias[n0 + 32 + col]; orow[32 + col] = fmaxf(v, 0.0f);
        v = c3[r] * si + bias[n0 + 48 + col]; orow[48 + col] = fmaxf(v, 0.0f);
    }
}

// ---------------------------------------------------------------- GEMM layer 2
// out[M,64] = (A[M,128] * sout[row]) @ B[128,64]; one wave per 16x64 tile (full N)
__global__ void gemm2_kernel(const float* __restrict__ A, const float* __restrict__ sout,
                             const float* __restrict__ B, float* __restrict__ out) {
    const int K = NF, NB = NH;
    int m0 = blockIdx.x * 16;
    int lane = threadIdx.x;
    int lr   = lane & 15;
    int kh   = (lane >> 4) << 1;

    v8f c0 = {}, c1 = {}, c2 = {}, c3 = {};
    const float* arow = A + (long long)(m0 + lr) * K;
    float sa = sout[m0 + lr];

    #pragma unroll 4
    for (int k = 0; k < K; k += 4) {
        v2f a;
        a.x = arow[k + kh] * sa;
        a.y = arow[k + kh + 1] * sa;
        const float* b0 = B + (long long)(k + kh) * NB + lr;
        v2f bb0, bb1, bb2, bb3;
        bb0.x = b0[0];  bb0.y = b0[NB];
        bb1.x = b0[16]; bb1.y = b0[NB + 16];
        bb2.x = b0[32]; bb2.y = b0[NB + 32];
        bb3.x = b0[48]; bb3.y = b0[NB + 48];
        c0 = __builtin_amdgcn_wmma_f32_16x16x4_f32(false, a, false, bb0, (short)0, c0, false, false);
        c1 = __builtin_amdgcn_wmma_f32_16x16x4_f32(false, a, false, bb1, (short)0, c1, false, false);
        c2 = __builtin_amdgcn_wmma_f32_16x16x4_f32(false, a, false, bb2, (short)0, c2, false, false);
        c3 = __builtin_amdgcn_wmma_f32_16x16x4_f32(false, a, false, bb3, (short)0, c3, false, false);
    }

    int rbase = m0 + ((lane >> 4) << 3);
    int col   = lane & 15;
    #pragma unroll
    for (int r = 0; r < 8; ++r) {
        float* orow = out + (long long)(rbase + r) * NB;
        orow[col]      = c0[r];
        orow[16 + col] = c1[r];
        orow[32 + col] = c2[r];
        orow[48 + col] = c3[r];
    }
}

// --------------------------------- out = log_softmax(agg*sin[row] + b2), 64 cols/row
// one wave per row; each lane owns cols {lane, lane+32}
__global__ void final_kernel(const float* __restrict__ agg, const float* __restrict__ sin_,
                             const float* __restrict__ bias, float* __restrict__ out) {
    int wid  = threadIdx.x >> 5;
    int lane = threadIdx.x & 31;
    int row  = blockIdx.x * 8 + wid;
    float si = sin_[row];
    const float* a = agg + (long long)row * NH;
    float v0 = a[lane]      * si + bias[lane];
    float v1 = a[lane + 32] * si + bias[lane + 32];

    float m = fmaxf(v0, v1);
    #pragma unroll
    for (int o = 16; o > 0; o >>= 1) m = fmaxf(m, __shfl_xor(m, o, 32));

    float s = expf(v0 - m) + expf(v1 - m);
    #pragma unroll
    for (int o = 16; o > 0; o >>= 1) s += __shfl_xor(s, o, 32);

    float l = m + logf(s);
    float* orow = out + (long long)row * NH;
    orow[lane]      = v0 - l;
    orow[lane + 32] = v1 - l;
}

extern "C" void kernel_launch(void* const* d_in, const int* in_sizes, int n_in,
                              void* d_out, int out_size, void* d_ws, size_t ws_size,
                              hipStream_t stream) {
    const float* feat = (const float*)d_in[0];
    const int*   src  = (const int*)d_in[1];
    const int*   dst  = (const int*)d_in[2];
    const float* W1   = (const float*)d_in[3];
    const float* b1   = (const float*)d_in[4];
    const float* W2   = (const float*)d_in[5];
    const float* b2   = (const float*)d_in[6];
    float* out = (float*)d_out;
    float* ws  = (float*)d_ws;

    const int DPAD = 50176;                 // padded degree-buffer length
    float* sOut = ws;                       // deg_out accum -> scale
    float* sIn  = ws + DPAD;                // deg_in accum -> scale
    float* bufA = ws + 2 * DPAD;            // h1 (scaled feat), later x   [50000*128]
    float* bufB = bufA + NNODES * NF;       // agg1                        [50000*128]
    float* bufC = bufB + NNODES * NF;       // h2                          [50000*64]
    float* bufD = bufC + NNODES * NH;       // agg2                        [50000*64]

    // reset accumulators every call (graph-capture safe)
    hipMemsetAsync(sOut, 0, 2 * DPAD * sizeof(float), stream);
    hipMemsetAsync(bufB, 0, (size_t)NNODES * NF * sizeof(float), stream);
    hipMemsetAsync(bufD, 0, (size_t)NNODES * NH * sizeof(float), stream);

    // degrees and D^{-1/2}
    deg_kernel<<<(NEDGES + 255) / 256, 256, 0, stream>>>(src, dst, sOut, sIn, NEDGES);
    rscale_kernel<<<(2 * DPAD + 255) / 256, 256, 0, stream>>>(sOut, 2 * DPAD);

    // ----- layer 1: scale -> aggregate(128) -> GEMM(+scale_in,+b1,relu)
    scale_rows_kernel<<<(NNODES * 32 + 255) / 256, 256, 0, stream>>>(feat, sOut, bufA, NNODES);
    scatter128_kernel<<<(NEDGES * 32 + 255) / 256, 256, 0, stream>>>(bufA, src, dst, bufB, NEDGES);
    gemm1_kernel<<<(NNODES / 16) * 2, 32, 0, stream>>>(bufB, W1, b1, sIn, bufA);

    // ----- layer 2: (x*scale_out) @ W2 -> aggregate(64) -> scale_in + b2 -> log_softmax
    gemm2_kernel<<<NNODES / 16, 32, 0, stream>>>(bufA, sOut, W2, bufC);
    scatter64_kernel<<<(NEDGES * 16 + 255) / 256, 256, 0, stream>>>(bufC, src, dst, bufD, NEDGES);
    final_kernel<<<NNODES / 8, 256, 0, stream>>>(bufD, sIn, b2, out);
}